// Attention_50190987821384
// MI455X (gfx1250) — compile-verified
//
#include <hip/hip_runtime.h>
#include <hip/hip_bf16.h>
#include <math.h>

// CDNA5 (gfx1250) flash-attention, fp32 WMMA (V_WMMA_F32_16X16X4_F32).
// B=8, T=2048, D=1024.  grid = B * (T/16) workgroups of 256 threads (8 wave32).
// Per key-tile of 16: each wave does 32 QK^T wmma (D-reduction slice of 128)
// and 32 PV wmma (D-column slice of 128).  Online softmax in wave 0.
// K/V tiles staged with CDNA5 async global->LDS loads (ASYNCcnt) when the
// toolchain exposes the builtin, else synchronous b128 copies.

typedef __attribute__((ext_vector_type(2))) float v2f;
typedef __attribute__((ext_vector_type(4))) float v4f;
typedef __attribute__((ext_vector_type(8))) float v8f;
typedef __attribute__((ext_vector_type(4))) int   v4i;

#define T_SEQ   2048
#define D_DIM   1024
#define QTILE   16
#define KTILE   16
#define NWAVE   8
#define DSLICE  128              // D per wave
#define KROW    1028             // padded LDS row stride (floats): conflict-free
#define PROW    20               // padded P/x row stride (floats)
#define INV_SCALE 0.03125f       // 1/sqrt(1024)
#define BIG     1.0e10f          // reference INF

#if __has_builtin(__builtin_amdgcn_global_load_async_to_lds_b128)
#define ASYNC_STAGING 1
#else
#define ASYNC_STAGING 0
#endif

typedef __attribute__((address_space(1))) const void gconst_t;  // global AS void
typedef __attribute__((address_space(3))) void lds_void_t;      // LDS AS void
typedef __attribute__((address_space(1))) v4i g_v4i;            // global AS int4
typedef __attribute__((address_space(3))) v4i l_v4i;            // LDS AS int4

// copy 16 bytes global -> LDS; async (ASYNCcnt-tracked) if available
static __device__ __forceinline__ void stage16(const float* g, float* l) {
#if ASYNC_STAGING
  g_v4i* gp = (g_v4i*)(gconst_t*)g;      // addrspacecast, then retype (drop const)
  l_v4i* lp = (l_v4i*)(lds_void_t*)l;
  __builtin_amdgcn_global_load_async_to_lds_b128(gp, lp, /*offset=*/0, /*cpol=*/0);
#else
  *(v4f*)l = *(const v4f*)g;
#endif
}

static __device__ __forceinline__ void stage_fence() {
#if ASYNC_STAGING
#if __has_builtin(__builtin_amdgcn_s_wait_asynccnt)
  __builtin_amdgcn_s_wait_asynccnt(0);
#else
  asm volatile("s_wait_asynccnt 0" ::: "memory");
#endif
#endif
}

__global__ __launch_bounds__(256) void fa_fp32_wmma_kernel(
    const float* __restrict__ Q, const float* __restrict__ K,
    const float* __restrict__ V, const float* __restrict__ Msk,
    float* __restrict__ O)
{
  extern __shared__ float smem[];
  float* kbuf = smem;                       // 16 * 1028
  float* vbuf = kbuf + 16 * KROW;           // 16 * 1028
  float* part = vbuf + 16 * KROW;           // 8 waves * 256 (v8f partial S tiles)
  float* xP   = part + NWAVE * 256;         // 16 * 20  (scores then probs)
  float* cfac = xP   + 16 * PROW;           // 16  per-row rescale factor
  float* lbuf = cfac + 16;                  // 16  per-row running denom

  const int tid  = threadIdx.x;
  const int wave = tid >> 5;
  const int lane = tid & 31;
  const int lq   = lane & 15;               // column / M index within half-wave
  const int hi   = lane >> 4;               // which half of the wave
  const int b    = blockIdx.x >> 7;         // T/16 = 128 query tiles per batch
  const int qt   = blockIdx.x & 127;
  const int q0   = qt * QTILE;
  const int d0w  = wave * DSLICE;

  const float* Qb = Q + (size_t)b * T_SEQ * D_DIM;
  const float* Kb = K + (size_t)b * T_SEQ * D_DIM;
  const float* Vb = V + (size_t)b * T_SEQ * D_DIM;
  const float* Mb = Msk + (size_t)b * T_SEQ;
  float*       Ob = O + (size_t)b * T_SEQ * D_DIM;

  // ---- preload this wave's Q slice as WMMA A operands (32 x v2f = 64 VGPRs)
  // A layout (f32 16x4): lanes 0-15 hold K=0,1 ; lanes 16-31 hold K=2,3.
  v2f aQ[32];
  {
    const float* qrow = Qb + (size_t)(q0 + lq) * D_DIM + d0w + hi * 2;
    #pragma unroll
    for (int j = 0; j < 32; ++j)
      aQ[j] = *(const v2f*)(qrow + j * 4);
  }

  // ---- output accumulator: 16 rows x 128 cols = 8 tiles of 16x16 f32
  v8f acc[8];
  {
    v8f z = {};
    #pragma unroll
    for (int i = 0; i < 8; ++i) acc[i] = z;
  }

  float m_run = -1.0e30f;   // running max (scaled logits), wave0 lanes 0-15
  float l_run = 0.0f;       // running denom,               wave0 lanes 0-15
  if (tid < 16) { cfac[tid] = 1.0f; lbuf[tid] = 0.0f; }

  const int nkt = qt + 1;                   // causal: key tiles 0..qt
  for (int kt = 0; kt < nkt; ++kt) {
    const int k0 = kt * KTILE;

    // ---- stage K,V tiles (16 x 1024 each) into LDS, 16B granules.
    __syncthreads();                        // previous iter done with kbuf/vbuf
    #pragma unroll
    for (int i = 0; i < 16; ++i) {
      const int row = i;                    // idx = i*256 + tid; row = idx>>8
      const int col = tid * 4;              // (idx & 255) * 4
      stage16(Kb + (size_t)(k0 + row) * D_DIM + col, kbuf + row * KROW + col);
      stage16(Vb + (size_t)(k0 + row) * D_DIM + col, vbuf + row * KROW + col);
    }
    // hint-prefetch next tile into L2 while we compute this one
    if (kt + 1 < nkt) {
      const float* nk = Kb + (size_t)(k0 + KTILE) * D_DIM;
      const float* nv = Vb + (size_t)(k0 + KTILE) * D_DIM;
      __builtin_prefetch(nk + tid * 64, 0, 0);
      __builtin_prefetch(nv + tid * 64, 0, 0);
    }
    stage_fence();                          // wave's async copies landed in LDS
    __syncthreads();

    // ---- partial S = Q_slice * K_slice^T   (32 wmma over this wave's 128 d)
    {
      v8f s = {};
      const float* kb = kbuf + lq * KROW + d0w + hi * 2;   // B: K^T[d][k]
      #pragma unroll
      for (int j = 0; j < 32; ++j) {
        v2f bk = *(const v2f*)(kb + j * 4);
        s = __builtin_amdgcn_wmma_f32_16x16x4_f32(
                false, aQ[j], false, bk, (short)0, s, false, false);
      }
      float* pw = part + wave * 256 + lane;
      #pragma unroll
      for (int r = 0; r < 8; ++r) pw[r * 32] = s[r];
    }
    __syncthreads();

    // ---- reduce 8 partials, apply causal + padding mask, scale
    {
      const int q = tid >> 4, kk = tid & 15;
      const int r = q & 7, li = kk + ((q >> 3) << 4);
      float s = 0.0f;
      #pragma unroll
      for (int w = 0; w < NWAVE; ++w) s += part[w * 256 + r * 32 + li];
      const int kg = k0 + kk;
      if (kg > q0 + q) s -= BIG;                 // causal (triu(1) * INF)
      s -= (1.0f - Mb[kg]) * BIG;                // key padding mask
      xP[q * PROW + kk] = s * INV_SCALE;         // scale AFTER masking
    }
    __syncthreads();

    // ---- online softmax: lane t < 16 owns query row t
    if (tid < 16) {
      float* xr = xP + tid * PROW;
      float mn = m_run;
      #pragma unroll
      for (int kk = 0; kk < 16; ++kk) mn = fmaxf(mn, xr[kk]);
      const float c = __expf(m_run - mn);
      float ps = 0.0f;
      #pragma unroll
      for (int kk = 0; kk < 16; ++kk) {
        const float p = __expf(xr[kk] - mn);
        xr[kk] = p;
        ps += p;
      }
      m_run = mn;
      l_run = l_run * c + ps;
      cfac[tid] = c;
      lbuf[tid] = l_run;
    }
    __syncthreads();

    // ---- rescale accumulators, then O += P * V_slice  (32 wmma)
    {
      float f[8];
      #pragma unroll
      for (int r = 0; r < 8; ++r) f[r] = cfac[r + hi * 8];   // row M = r + hi*8
      #pragma unroll
      for (int ot = 0; ot < 8; ++ot) {
        #pragma unroll
        for (int r = 0; r < 8; ++r) acc[ot][r] *= f[r];
      }
      #pragma unroll
      for (int jk = 0; jk < 4; ++jk) {          // key chunks of 4
        const v2f ap = *(const v2f*)(xP + lq * PROW + jk * 4 + hi * 2);
        const int kr = jk * 4 + hi * 2;
        #pragma unroll
        for (int ot = 0; ot < 8; ++ot) {
          const int dc = d0w + ot * 16 + lq;
          v2f bv;
          bv.x = vbuf[kr * KROW + dc];          // B: V[k][d], rows kr,kr+1
          bv.y = vbuf[(kr + 1) * KROW + dc];
          acc[ot] = __builtin_amdgcn_wmma_f32_16x16x4_f32(
                        false, ap, false, bv, (short)0, acc[ot], false, false);
        }
      }
    }
  }

  // ---- normalize by running denom and write out
  __syncthreads();
  {
    float linv[8];
    #pragma unroll
    for (int r = 0; r < 8; ++r) linv[r] = 1.0f / lbuf[r + hi * 8];
    #pragma unroll
    for (int ot = 0; ot < 8; ++ot) {
      const int dc = d0w + ot * 16 + lq;
      #pragma unroll
      for (int r = 0; r < 8; ++r) {
        const int row = q0 + r + hi * 8;
        Ob[(size_t)row * D_DIM + dc] = acc[ot][r] * linv[r];
      }
    }
  }
}

extern "C" void kernel_launch(void* const* d_in, const int* in_sizes, int n_in,
                              void* d_out, int out_size, void* d_ws, size_t ws_size,
                              hipStream_t stream) {
  const float* Q   = (const float*)d_in[0];
  const float* K   = (const float*)d_in[1];
  const float* V   = (const float*)d_in[2];
  const float* Msk = (const float*)d_in[3];
  float* O = (float*)d_out;

  const int B = in_sizes[3] / T_SEQ;            // mask is [B, T]
  const int qtiles = T_SEQ / QTILE;             // 128

  // dynamic LDS: K tile + V tile + partials + P tile + factors
  const size_t shmem =
      (size_t)(16 * KROW * 2 + NWAVE * 256 + 16 * PROW + 16 + 16) * sizeof(float);

  dim3 grid(B * qtiles);
  dim3 block(256);
  fa_fp32_wmma_kernel<<<grid, block, shmem, stream>>>(Q, K, V, Msk, O);
}